// OverallLoss_36009005810130
// MI455X (gfx1250) — compile-verified
//
#include <hip/hip_runtime.h>
#include <math.h>

typedef __attribute__((ext_vector_type(2))) float v2f;
typedef __attribute__((ext_vector_type(8))) float v8f;

#define FRAME_W 1920.0f
#define FRAME_H 1080.0f

// uniform broadcast from a specific lane: v_readlane_b32 (no LDS round-trip)
__device__ __forceinline__ float read_lane_f32(float v, int srcLane) {
  return __int_as_float(__builtin_amdgcn_readlane(__float_as_int(v), srcLane));
}

// 3x3 inverse via adjugate (general, faithful to jnp.linalg.inv on these inputs)
__device__ __forceinline__ void inv3x3(const float M[3][3], float Mi[3][3]) {
  float c00 =  M[1][1]*M[2][2] - M[1][2]*M[2][1];
  float c01 = -(M[1][0]*M[2][2] - M[1][2]*M[2][0]);
  float c02 =  M[1][0]*M[2][1] - M[1][1]*M[2][0];
  float c10 = -(M[0][1]*M[2][2] - M[0][2]*M[2][1]);
  float c11 =  M[0][0]*M[2][2] - M[0][2]*M[2][0];
  float c12 = -(M[0][0]*M[2][1] - M[0][1]*M[2][0]);
  float c20 =  M[0][1]*M[1][2] - M[0][2]*M[1][1];
  float c21 = -(M[0][0]*M[1][2] - M[0][2]*M[1][0]);
  float c22 =  M[0][0]*M[1][1] - M[0][1]*M[1][0];
  float id = 1.0f / (M[0][0]*c00 + M[0][1]*c01 + M[0][2]*c02);
  Mi[0][0]=c00*id; Mi[0][1]=c10*id; Mi[0][2]=c20*id;
  Mi[1][0]=c01*id; Mi[1][1]=c11*id; Mi[1][2]=c21*id;
  Mi[2][0]=c02*id; Mi[2][1]=c12*id; Mi[2][2]=c22*id;
}

__global__ __launch_bounds__(32) void overall_loss_kernel(
    const float* __restrict__ Rotation, const float* __restrict__ center,
    const float* __restrict__ bbox,     const float* __restrict__ bbox_3d,
    const float* __restrict__ ER_p,     const float* __restrict__ et_p,
    const float* __restrict__ K_p,      const float* __restrict__ kx_p,
    const float* __restrict__ ky_p,     const float* __restrict__ beta_p,
    const float* __restrict__ t_p,      const float* __restrict__ Rq,
    const float* __restrict__ s_p,      const float* __restrict__ s_cls,
    float* __restrict__ out)
{
  const int  lane = threadIdx.x;
  const int  m    = lane & 15;        // row (A) or column (B/D) index
  const bool hi   = lane >= 16;       // upper lane half -> K = 2,3

  // ---- uniform loads (scalarized by compiler into s_load) -----------------
  float ERm[3][3], Km[3][3], Rot[3][3];
  #pragma unroll
  for (int i = 0; i < 3; ++i)
    #pragma unroll
    for (int j = 0; j < 3; ++j) {
      ERm[i][j] = ER_p[3*i+j];
      Km [i][j] = K_p [3*i+j];
      Rot[i][j] = Rotation[3*i+j];
    }
  float etv[3] = { et_p[0], et_p[1], et_p[2] };
  float tv [3] = { t_p[0],  t_p[1],  t_p[2]  };
  float sv [3] = { s_p[0],  s_p[1],  s_p[2]  };
  float kxv = kx_p[0], kyv = ky_p[0], betav = beta_p[0];

  // ---- quaternion -> R_mat -------------------------------------------------
  float q0 = Rq[0], q1 = Rq[1], q2 = Rq[2], q3 = Rq[3];
  float qn = sqrtf(q0*q0 + q1*q1 + q2*q2 + q3*q3);
  float x0 = q0/qn, x1 = q1/qn, x2 = q2/qn, x3 = q3/qn;
  float Rm[3][3] = {
    {1.f-2.f*(x1*x1+x2*x2), 2.f*(x0*x1-x2*x3),     2.f*(x0*x2+x1*x3)},
    {2.f*(x0*x1+x2*x3),     1.f-2.f*(x0*x0+x2*x2), 2.f*(x1*x2-x0*x3)},
    {2.f*(x0*x2-x1*x3),     2.f*(x1*x2+x0*x3),     1.f-2.f*(x0*x0+x1*x1)}};

  // ---- ext_inv of [[ER,0],[et,1]] = [[ERi,0],[-et@ERi,1]] ------------------
  float ERi[3][3];
  inv3x3(ERm, ERi);
  float wrow[3];
  #pragma unroll
  for (int j = 0; j < 3; ++j)
    wrow[j] = -(etv[0]*ERi[0][j] + etv[1]*ERi[1][j] + etv[2]*ERi[2][j]);

  // ---- A4 = diag(s,1) @ [[R^T,0],[t,1]] @ ext_inv   (last col = e3, elided)
  float A4[4][3];
  #pragma unroll
  for (int i = 0; i < 3; ++i)
    #pragma unroll
    for (int j = 0; j < 3; ++j) {
      float acc = 0.f;
      #pragma unroll
      for (int k = 0; k < 3; ++k) acc += Rm[k][i]*ERi[k][j];
      A4[i][j] = sv[i]*acc;
    }
  #pragma unroll
  for (int j = 0; j < 3; ++j)
    A4[3][j] = tv[0]*ERi[0][j] + tv[1]*ERi[1][j] + tv[2]*ERi[2][j] + wrow[j];

  // ---- fold K: Bm col0 = A4·Krow0 (x-num), col1 = A4·Krow1 (y-num), col2 = z
  float Bm[4][3];
  #pragma unroll
  for (int k = 0; k < 4; ++k) {
    Bm[k][0] = A4[k][0]*Km[0][0] + A4[k][1]*Km[0][1] + A4[k][2]*Km[0][2];
    Bm[k][1] = A4[k][0]*Km[1][0] + A4[k][1]*Km[1][1] + A4[k][2]*Km[1][2];
    Bm[k][2] = A4[k][2];
  }

  // ---- WMMA operands -------------------------------------------------------
  // A (16x4 f32): lane m<8 = corner hom [c0,c1,c2,1]; rows 8..15 zero.
  float c0 = ((m>>2)&1) ? bbox_3d[1] : bbox_3d[0];
  float c1 = ((m>>1)&1) ? bbox_3d[3] : bbox_3d[2];
  float c2 = ( m     &1) ? bbox_3d[5] : bbox_3d[4];
  bool  valid = (m < 8);
  float h0 = valid ? c0 : 0.f, h1 = valid ? c1 : 0.f;
  float h2 = valid ? c2 : 0.f, h3 = valid ? 1.f : 0.f;
  v2f a;
  a.x = hi ? h2 : h0;   // K = kb
  a.y = hi ? h3 : h1;   // K = kb+1

  // B (4x16 f32): N = lane%16, lanes 0-15 hold K=0,1 ; lanes 16-31 hold K=2,3
  float r0c0 = hi ? Bm[2][0] : Bm[0][0];
  float r0c1 = hi ? Bm[2][1] : Bm[0][1];
  float r0c2 = hi ? Bm[2][2] : Bm[0][2];
  float r1c0 = hi ? Bm[3][0] : Bm[1][0];
  float r1c1 = hi ? Bm[3][1] : Bm[1][1];
  float r1c2 = hi ? Bm[3][2] : Bm[1][2];
  v2f b;
  b.x = (m==0) ? r0c0 : (m==1) ? r0c1 : (m==2) ? r0c2 : 0.f;
  b.y = (m==0) ? r1c0 : (m==1) ? r1c1 : (m==2) ? r1c2 : 0.f;

  v8f cacc = {};
  // D = A(16x4) x B(4x16): one matrix op does all 8 corner projections
  v8f d = __builtin_amdgcn_wmma_f32_16x16x4_f32(
      false, a, false, b, (short)0, cacc, false, false);

  // ---- per-column reduction: lane0 = x-num, lane1 = y-num, lane2 = z ------
  // z per corner is the same for all lanes after broadcast -> v_readlane
  float rmin =  3.0e38f, rmax = -3.0e38f;
  #pragma unroll
  for (int i = 0; i < 8; ++i) {
    float num = d[i];
    float z   = read_lane_f32(d[i], 2);   // z of corner i (column 2), uniform
    float r   = num / z;
    rmin = fminf(rmin, r);
    rmax = fmaxf(rmax, r);
  }
  float minx = read_lane_f32(rmin, 0), maxx = read_lane_f32(rmax, 0);
  float miny = read_lane_f32(rmin, 1), maxy = read_lane_f32(rmax, 1);

  // ---- scale_bbox_loss -----------------------------------------------------
  float sbl = 0.25f * ( fabsf(minx/FRAME_W - bbox[0])
                      + fabsf(maxx/FRAME_W - bbox[1])
                      + fabsf(miny/FRAME_H - bbox[2])
                      + fabsf(maxy/FRAME_H - bbox[3]) );

  // ---- world_center_loss ---------------------------------------------------
  float Ki[3][3];
  inv3x3(Km, Ki);
  float ccv[3] = { betav*kxv*FRAME_W, betav*kyv*FRAME_H, betav };
  float cc[3];
  #pragma unroll
  for (int i = 0; i < 3; ++i)
    cc[i] = Ki[i][0]*ccv[0] + Ki[i][1]*ccv[1] + Ki[i][2]*ccv[2];
  float wcl = 0.f;
  #pragma unroll
  for (int j = 0; j < 3; ++j) {
    float world_j = cc[0]*ERm[0][j] + cc[1]*ERm[1][j] + cc[2]*ERm[2][j] + etv[j];
    wcl += fabsf(world_j - tv[j]);
  }
  wcl *= (1.0f/3.0f);

  // ---- pixel_center_loss ---------------------------------------------------
  float pcl = 0.5f * (fabsf(kxv - center[0]) + fabsf(kyv - center[1]));

  // ---- rotation_loss -------------------------------------------------------
  float rl = 0.f;
  #pragma unroll
  for (int i = 0; i < 3; ++i)
    #pragma unroll
    for (int j = 0; j < 3; ++j) {
      float er = ERm[i][0]*Rm[0][j] + ERm[i][1]*Rm[1][j] + ERm[i][2]*Rm[2][j];
      float dd = Rot[i][j] - er;
      rl += dd*dd;
    }
  rl *= (1.0f/9.0f);

  // ---- scale_cls_loss ------------------------------------------------------
  float scl = (fabsf(sv[0]-s_cls[0]) + fabsf(sv[1]-s_cls[1]) + fabsf(sv[2]-s_cls[2]))
              * (1.0f/3.0f);

  float overall = 0.2f * (pcl + wcl + rl + sbl + scl);

  if (lane == 0) {
    out[0] = overall;
    out[1] = pcl;
    out[2] = wcl;
    out[3] = rl;
    out[4] = sbl;
    out[5] = scl;
  }
}

extern "C" void kernel_launch(void* const* d_in, const int* in_sizes, int n_in,
                              void* d_out, int out_size, void* d_ws, size_t ws_size,
                              hipStream_t stream) {
  (void)in_sizes; (void)n_in; (void)d_ws; (void)ws_size; (void)out_size;
  const float* Rotation = (const float*)d_in[0];
  const float* center   = (const float*)d_in[1];
  const float* bbox     = (const float*)d_in[2];
  const float* bbox_3d  = (const float*)d_in[3];
  const float* ER       = (const float*)d_in[4];
  const float* et       = (const float*)d_in[5];
  const float* K        = (const float*)d_in[6];
  const float* kx       = (const float*)d_in[7];
  const float* ky       = (const float*)d_in[8];
  const float* beta     = (const float*)d_in[9];
  const float* t        = (const float*)d_in[10];
  const float* Rq       = (const float*)d_in[11];
  const float* s        = (const float*)d_in[12];
  const float* s_cls    = (const float*)d_in[13];
  float* out = (float*)d_out;

  overall_loss_kernel<<<1, 32, 0, stream>>>(Rotation, center, bbox, bbox_3d,
                                            ER, et, K, kx, ky, beta, t, Rq,
                                            s, s_cls, out);
}